// Attention_14035953123627
// MI455X (gfx1250) — compile-verified
//
#include <hip/hip_runtime.h>
#include <hip/hip_bf16.h>
#include <math.h>

typedef float v2f __attribute__((ext_vector_type(2)));
typedef float v8f __attribute__((ext_vector_type(8)));

#define WMMA4(a, b, c) \
    __builtin_amdgcn_wmma_f32_16x16x4_f32(false, (a), false, (b), (short)0, (c), false, false)

// ---------------------------------------------------------------------------
// Problem constants (from reference)
// ---------------------------------------------------------------------------
#define BATCH 4
#define NPIX  16384     // 128*128
#define CH    256
#define NH    8
#define HD    32
#define TQ    16384
#define NKV   1024      // (128/4)^2 per batch
#define TOK   4096      // BATCH * NKV
#define KCONV 4096      // 16 patch pixels * 256 ch
#define ATTN_SCALE 0.17677669529663687f   // 1/sqrt(32)

// ---------------------------------------------------------------------------
// Generic fp32 WMMA GEMM:  C[M,N] = A[M,K] @ W[K,N] (+bias)
// block = 128 threads = 4 waves (2x2), block tile 64x64,
// wave tile 32x32 (4 accumulators, 2x2 fragment reuse), K-chunk 32
// ---------------------------------------------------------------------------
__global__ __launch_bounds__(128)
void gemm_f32_kernel(const float* __restrict__ A, const float* __restrict__ W,
                     const float* __restrict__ bias, float* __restrict__ C,
                     int M, int N, int K) {
    __shared__ float As[64][33];
    __shared__ float Ws[32][68];

    const int tid  = threadIdx.x;
    const int wave = tid >> 5;
    const int lane = tid & 31;
    const int half = lane >> 4;
    const int l    = lane & 15;
    const int wm   = (wave >> 1) * 32;
    const int wn   = (wave & 1) * 32;
    const int bm   = blockIdx.y * 64;
    const int bn   = blockIdx.x * 64;

    v8f acc00 = {}, acc01 = {}, acc10 = {}, acc11 = {};

    for (int k0 = 0; k0 < K; k0 += 32) {
        #pragma unroll
        for (int i = 0; i < 4; ++i) {
            int idx = tid * 4 + i;          // 0..511 float4 slots
            int ar  = idx >> 3;             // 0..63
            int ac  = (idx & 7) * 4;        // 0..28
            float4 av = *(const float4*)(A + (size_t)(bm + ar) * K + k0 + ac);
            As[ar][ac + 0] = av.x; As[ar][ac + 1] = av.y;
            As[ar][ac + 2] = av.z; As[ar][ac + 3] = av.w;
            int wr = idx >> 4;              // 0..31
            int wc = (idx & 15) * 4;        // 0..60
            float4 wv = *(const float4*)(W + (size_t)(k0 + wr) * N + bn + wc);
            Ws[wr][wc + 0] = wv.x; Ws[wr][wc + 1] = wv.y;
            Ws[wr][wc + 2] = wv.z; Ws[wr][wc + 3] = wv.w;
        }
        // prefetch next K-chunk into GL2 (global_prefetch_b8)
        if (k0 + 32 < K) {
            __builtin_prefetch(A + (size_t)(bm + (tid & 63)) * K + k0 + 32, 0, 1);
            if (tid < 32)
                __builtin_prefetch(W + (size_t)(k0 + 32 + tid) * N + bn, 0, 1);
        }
        __syncthreads();
        #pragma unroll
        for (int kk = 0; kk < 32; kk += 4) {
            v2f a0, a1, b0, b1;
            a0.x = As[wm + l][kk + 2 * half];
            a0.y = As[wm + l][kk + 2 * half + 1];
            a1.x = As[wm + 16 + l][kk + 2 * half];
            a1.y = As[wm + 16 + l][kk + 2 * half + 1];
            b0.x = Ws[kk + 2 * half][wn + l];
            b0.y = Ws[kk + 2 * half + 1][wn + l];
            b1.x = Ws[kk + 2 * half][wn + 16 + l];
            b1.y = Ws[kk + 2 * half + 1][wn + 16 + l];
            acc00 = WMMA4(a0, b0, acc00);
            acc01 = WMMA4(a0, b1, acc01);
            acc10 = WMMA4(a1, b0, acc10);
            acc11 = WMMA4(a1, b1, acc11);
        }
        __syncthreads();
    }

    const int col0 = bn + wn + l;
    const int col1 = col0 + 16;
    const float bv0 = bias ? bias[col0] : 0.0f;
    const float bv1 = bias ? bias[col1] : 0.0f;
    #pragma unroll
    for (int r = 0; r < 8; ++r) {
        int row0 = bm + wm + r + 8 * half;
        int row1 = row0 + 16;
        C[(size_t)row0 * N + col0] = acc00[r] + bv0;
        C[(size_t)row0 * N + col1] = acc01[r] + bv1;
        C[(size_t)row1 * N + col0] = acc10[r] + bv0;
        C[(size_t)row1 * N + col1] = acc11[r] + bv1;
    }
}

// ---------------------------------------------------------------------------
// Conv 4x4 stride 4 as im2col GEMM: y[TOK,256] = patches[TOK,4096] @ W'[4096,256] + b
// K index k = p*256 + ci, p = i*4 + j (pixel-major, channel-minor) so each
// 32-wide K-chunk stays inside one patch pixel -> contiguous float4 A loads.
// ---------------------------------------------------------------------------
__global__ __launch_bounds__(128)
void conv_gemm_kernel(const float* __restrict__ x, const float* __restrict__ srw,
                      const float* __restrict__ srb, float* __restrict__ y) {
    __shared__ float As[64][33];
    __shared__ float Ws[32][68];

    const int tid  = threadIdx.x;
    const int wave = tid >> 5;
    const int lane = tid & 31;
    const int half = lane >> 4;
    const int l    = lane & 15;
    const int wm   = (wave >> 1) * 32;
    const int wn   = (wave & 1) * 32;
    const int bm   = blockIdx.y * 64;       // token tile
    const int bn   = blockIdx.x * 64;       // out-channel tile

    v8f acc00 = {}, acc01 = {}, acc10 = {}, acc11 = {};

    for (int k0 = 0; k0 < KCONV; k0 += 32) {
        const int p   = k0 >> 8;            // patch pixel 0..15 (constant in chunk)
        const int pi  = p >> 2;
        const int pj  = p & 3;
        const int ci0 = k0 & 255;
        #pragma unroll
        for (int i = 0; i < 4; ++i) {
            int idx = tid * 4 + i;
            int ar  = idx >> 3;             // 0..63
            int ac  = (idx & 7) * 4;
            int m   = bm + ar;
            int b   = m >> 10;
            int rem = m & 1023;
            int oy  = rem >> 5;
            int ox  = rem & 31;
            size_t xa = ((size_t)b * NPIX + (size_t)(4 * oy + pi) * 128 + (4 * ox + pj)) * CH
                        + ci0 + ac;
            float4 av = *(const float4*)(x + xa);
            As[ar][ac + 0] = av.x; As[ar][ac + 1] = av.y;
            As[ar][ac + 2] = av.z; As[ar][ac + 3] = av.w;
            int wr = idx >> 4;              // 0..31 (k row)
            int wc = (idx & 15) * 4;        // 0..60 (n)
            int ci = ci0 + wr;
            #pragma unroll
            for (int qn = 0; qn < 4; ++qn) {
                Ws[wr][wc + qn] = srw[(size_t)(bn + wc + qn) * KCONV + ci * 16 + pi * 4 + pj];
            }
        }
        __syncthreads();
        #pragma unroll
        for (int kk = 0; kk < 32; kk += 4) {
            v2f a0, a1, b0, b1;
            a0.x = As[wm + l][kk + 2 * half];
            a0.y = As[wm + l][kk + 2 * half + 1];
            a1.x = As[wm + 16 + l][kk + 2 * half];
            a1.y = As[wm + 16 + l][kk + 2 * half + 1];
            b0.x = Ws[kk + 2 * half][wn + l];
            b0.y = Ws[kk + 2 * half + 1][wn + l];
            b1.x = Ws[kk + 2 * half][wn + 16 + l];
            b1.y = Ws[kk + 2 * half + 1][wn + 16 + l];
            acc00 = WMMA4(a0, b0, acc00);
            acc01 = WMMA4(a0, b1, acc01);
            acc10 = WMMA4(a1, b0, acc10);
            acc11 = WMMA4(a1, b1, acc11);
        }
        __syncthreads();
    }

    const int col0 = bn + wn + l;
    const int col1 = col0 + 16;
    const float bv0 = srb[col0];
    const float bv1 = srb[col1];
    #pragma unroll
    for (int r = 0; r < 8; ++r) {
        int row0 = bm + wm + r + 8 * half;
        int row1 = row0 + 16;
        y[(size_t)row0 * CH + col0] = acc00[r] + bv0;
        y[(size_t)row0 * CH + col1] = acc01[r] + bv1;
        y[(size_t)row1 * CH + col0] = acc10[r] + bv0;
        y[(size_t)row1 * CH + col1] = acc11[r] + bv1;
    }
}

// ---------------------------------------------------------------------------
// LayerNorm over C=256 per token
// ---------------------------------------------------------------------------
__global__ __launch_bounds__(256)
void layernorm_kernel(const float* __restrict__ y, const float* __restrict__ g,
                      const float* __restrict__ b, float* __restrict__ out) {
    __shared__ float red[256];
    const int tid = threadIdx.x;
    const int row = blockIdx.x;
    float v = y[(size_t)row * CH + tid];

    red[tid] = v;
    __syncthreads();
    for (int s = 128; s > 0; s >>= 1) {
        if (tid < s) red[tid] += red[tid + s];
        __syncthreads();
    }
    float mu = red[0] * (1.0f / CH);
    __syncthreads();

    float d = v - mu;
    red[tid] = d * d;
    __syncthreads();
    for (int s = 128; s > 0; s >>= 1) {
        if (tid < s) red[tid] += red[tid + s];
        __syncthreads();
    }
    float var = red[0] * (1.0f / CH);

    out[(size_t)row * CH + tid] = d * rsqrtf(var + 1e-5f) * g[tid] + b[tid];
}

// ---------------------------------------------------------------------------
// Attention: one block per (16-query tile, head). 256 threads = 8 waves.
// Full 16x1024 score tile in LDS; per-wave KV staging; WMMA for QK^T and PV.
// Q fragments are hoisted into registers (invariant across key chunks).
// ---------------------------------------------------------------------------
#define S_STRIDE 1028   // 1024 + 4 pad (bank-offset rows)

__global__ __launch_bounds__(256)
void attn_kernel(const float* __restrict__ qh, const float* __restrict__ kvbuf,
                 const int* __restrict__ qlens, float* __restrict__ outp) {
    __shared__ float S[16][S_STRIDE];       // scores / probabilities
    __shared__ float Qs[16][32];
    __shared__ float KVs[8][16][33];        // per-wave K or V chunk staging
    __shared__ float Os[16][32];            // merged output tile
    __shared__ float prt[16][16];
    __shared__ float rowmax[16];
    __shared__ float rowsum[16];

    const int tid  = threadIdx.x;
    const int wave = tid >> 5;
    const int lane = tid & 31;
    const int half = lane >> 4;
    const int l    = lane & 15;
    const int row0 = blockIdx.x * 16;       // global query row base
    const int h    = blockIdx.y;            // head

    // batch of this tile (tiles never straddle batches: lengths are *16)
    int b = 0, off = 0;
    for (int i = 0; i < 4; ++i) {
        int L = qlens[i];
        if (row0 >= off + L) { off += L; b = i + 1; }
    }
    const size_t kvbase = (size_t)b * NKV * 512 + (size_t)h * HD;   // K at +0, V at +256

    // load Q tile, zero O tile
    {
        int r  = tid >> 4;
        int d0 = (tid & 15) * 2;
        Qs[r][d0]     = qh[(size_t)(row0 + r) * CH + h * HD + d0];
        Qs[r][d0 + 1] = qh[(size_t)(row0 + r) * CH + h * HD + d0 + 1];
        float* oz = &Os[0][0];
        oz[tid]       = 0.0f;
        oz[tid + 256] = 0.0f;
    }
    __syncthreads();

    // hoisted Q A-fragments: invariant across all key chunks
    v2f qa[8];
    #pragma unroll
    for (int kk = 0; kk < 8; ++kk) {
        qa[kk].x = Qs[l][kk * 4 + 2 * half];
        qa[kk].y = Qs[l][kk * 4 + 2 * half + 1];
    }

    // ---- phase 1: S = (Q @ K^T) * scale, wave w owns keys [w*128, w*128+128)
    for (int t = 0; t < 8; ++t) {
        const int kc0 = wave * 128 + t * 16;
        // stage K chunk [16 keys x 32 dims] (wave-local; DS ops are in-order)
        #pragma unroll
        for (int u = 0; u < 4; ++u) {
            int idx = lane * 4 + u;         // 0..127 float4 slots
            int key = idx >> 3;
            int d4  = (idx & 7) * 4;
            float4 kv4 = *(const float4*)(kvbuf + kvbase + (size_t)(kc0 + key) * 512 + d4);
            KVs[wave][key][d4 + 0] = kv4.x; KVs[wave][key][d4 + 1] = kv4.y;
            KVs[wave][key][d4 + 2] = kv4.z; KVs[wave][key][d4 + 3] = kv4.w;
        }
        v8f s = {};
        #pragma unroll
        for (int kk = 0; kk < 8; ++kk) {
            v2f bb;
            bb.x = KVs[wave][l][kk * 4 + 2 * half];       // B[k=d][n=key]
            bb.y = KVs[wave][l][kk * 4 + 2 * half + 1];
            s = WMMA4(qa[kk], bb, s);
        }
        #pragma unroll
        for (int r = 0; r < 8; ++r) {
            S[r + 8 * half][kc0 + l] = s[r] * ATTN_SCALE;
        }
    }
    __syncthreads();

    // ---- phase 2: row softmax (16 threads per row, 64 keys each)
    {
        const int srow = tid >> 4;
        const int sseg = tid & 15;
        const int base = sseg * 64;
        float mx = -3.4e38f;
        for (int i = 0; i < 64; ++i) mx = fmaxf(mx, S[srow][base + i]);
        prt[srow][sseg] = mx;
        __syncthreads();
        if (sseg == 0) {
            float m2 = prt[srow][0];
            for (int i = 1; i < 16; ++i) m2 = fmaxf(m2, prt[srow][i]);
            rowmax[srow] = m2;
        }
        __syncthreads();
        float rm = rowmax[srow];
        float ps = 0.0f;
        for (int i = 0; i < 64; ++i) {
            float e = __expf(S[srow][base + i] - rm);
            S[srow][base + i] = e;
            ps += e;
        }
        prt[srow][sseg] = ps;
        __syncthreads();
        if (sseg == 0) {
            float s2 = 0.0f;
            for (int i = 0; i < 16; ++i) s2 += prt[srow][i];
            rowsum[srow] = s2;
        }
        __syncthreads();
        float inv = 1.0f / rowsum[srow];
        for (int i = 0; i < 64; ++i) S[srow][base + i] *= inv;
    }
    __syncthreads();

    // ---- phase 3: O += P @ V over this wave's key range, merge via LDS atomics
    v8f o0 = {}, o1 = {};
    for (int t = 0; t < 8; ++t) {
        const int kc0 = wave * 128 + t * 16;
        #pragma unroll
        for (int u = 0; u < 4; ++u) {
            int idx = lane * 4 + u;
            int key = idx >> 3;
            int d4  = (idx & 7) * 4;
            float4 kv4 = *(const float4*)(kvbuf + kvbase + 256 + (size_t)(kc0 + key) * 512 + d4);
            KVs[wave][key][d4 + 0] = kv4.x; KVs[wave][key][d4 + 1] = kv4.y;
            KVs[wave][key][d4 + 2] = kv4.z; KVs[wave][key][d4 + 3] = kv4.w;
        }
        #pragma unroll
        for (int kk = 0; kk < 16; kk += 4) {
            v2f a, b0, b1;
            a.x  = S[l][kc0 + kk + 2 * half];
            a.y  = S[l][kc0 + kk + 2 * half + 1];
            b0.x = KVs[wave][kk + 2 * half][l];           // B[k=key][n=d], d 0..15
            b0.y = KVs[wave][kk + 2 * half + 1][l];
            b1.x = KVs[wave][kk + 2 * half][16 + l];      // d 16..31
            b1.y = KVs[wave][kk + 2 * half + 1][16 + l];
            o0 = WMMA4(a, b0, o0);
            o1 = WMMA4(a, b1, o1);
        }
    }
    #pragma unroll
    for (int r = 0; r < 8; ++r) {
        atomicAdd(&Os[r + 8 * half][l],      o0[r]);
        atomicAdd(&Os[r + 8 * half][16 + l], o1[r]);
    }
    __syncthreads();

    {
        int r  = tid >> 4;
        int d0 = (tid & 15) * 2;
        outp[(size_t)(row0 + r) * CH + h * HD + d0]     = Os[r][d0];
        outp[(size_t)(row0 + r) * CH + h * HD + d0 + 1] = Os[r][d0 + 1];
    }
}

// ---------------------------------------------------------------------------
// launcher
// ---------------------------------------------------------------------------
extern "C" void kernel_launch(void* const* d_in, const int* in_sizes, int n_in,
                              void* d_out, int out_size, void* d_ws, size_t ws_size,
                              hipStream_t stream) {
    (void)in_sizes; (void)n_in; (void)out_size; (void)ws_size;

    const float* x      = (const float*)d_in[0];
    const float* q      = (const float*)d_in[1];
    const int*   qlens  = (const int*)  d_in[4];
    const float* w_q    = (const float*)d_in[5];
    const float* w_kv   = (const float*)d_in[6];
    const float* sr_w   = (const float*)d_in[7];
    const float* sr_b   = (const float*)d_in[8];
    const float* ln_g   = (const float*)d_in[9];
    const float* ln_b   = (const float*)d_in[10];
    const float* proj_w = (const float*)d_in[11];
    const float* proj_b = (const float*)d_in[12];
    float* out = (float*)d_out;

    float* ws      = (float*)d_ws;
    float* xconv   = ws;                                // TOK*256   floats
    float* xln     = xconv + (size_t)TOK * CH;
    float* kvbuf   = xln   + (size_t)TOK * CH;          // TOK*512
    float* qhbuf   = kvbuf + (size_t)TOK * 512;         // TQ*256
    float* attnout = qhbuf + (size_t)TQ * CH;           // TQ*256

    // 1) q projection: [TQ,256] = q @ w_q
    gemm_f32_kernel<<<dim3(CH / 64, TQ / 64), 128, 0, stream>>>(
        q, w_q, nullptr, qhbuf, TQ, CH, CH);

    // 2) spatial reduction conv (im2col GEMM) + bias
    conv_gemm_kernel<<<dim3(CH / 64, TOK / 64), 128, 0, stream>>>(
        x, sr_w, sr_b, xconv);

    // 3) LayerNorm
    layernorm_kernel<<<dim3(TOK), 256, 0, stream>>>(xconv, ln_g, ln_b, xln);

    // 4) kv projection: [TOK,512] = xln @ w_kv
    gemm_f32_kernel<<<dim3(512 / 64, TOK / 64), 128, 0, stream>>>(
        xln, w_kv, nullptr, kvbuf, TOK, 512, CH);

    // 5) ragged attention
    attn_kernel<<<dim3(TQ / 16, NH), 256, 0, stream>>>(
        qhbuf, kvbuf, qlens, attnout);

    // 6) output projection + bias
    gemm_f32_kernel<<<dim3(CH / 64, TQ / 64), 128, 0, stream>>>(
        attnout, proj_w, proj_b, out, TQ, CH, CH);
}